// GCMCLayer_73796128079917
// MI455X (gfx1250) — compile-verified
//
#include <hip/hip_runtime.h>

typedef __attribute__((ext_vector_type(2))) float v2f;
typedef __attribute__((ext_vector_type(8))) float v8f;
typedef __attribute__((ext_vector_type(4))) int   v4i;

#define F_DIM 128
#define EFF_DIM 128
#define OUT_DIM 64
#define KT 32   // K chunk staged in LDS
#define SA_STRIDE 36  // 16B-aligned rows for async b128, conflict-free frag reads

#if __has_builtin(__builtin_amdgcn_global_load_async_to_lds_b128)
#define HAS_ASYNC_LDS 1
#else
#define HAS_ASYNC_LDS 0
#endif

#if HAS_ASYNC_LDS
typedef __attribute__((address_space(1))) v4i* as1_i4p;
typedef __attribute__((address_space(3))) v4i* as3_i4p;
__device__ __forceinline__ void async_wait0() {
#if __has_builtin(__builtin_amdgcn_s_wait_asynccnt)
    __builtin_amdgcn_s_wait_asynccnt(0);
#else
    asm volatile("s_wait_asynccnt 0" ::: "memory");
#endif
}
#endif

// ---------------------------------------------------------------------------
// W[r] = sum_b att[r,b] * basis[b]   -> [R, F, EFF]
// ---------------------------------------------------------------------------
__global__ void build_w_kernel(const float* __restrict__ att,
                               const float* __restrict__ basis,
                               float* __restrict__ W,
                               int R, int B, int FE /* F*EFF */) {
    int idx = blockIdx.x * blockDim.x + threadIdx.x;
    int total = R * FE;
    if (idx >= total) return;
    int r  = idx / FE;
    int fe = idx - r * FE;
    float acc = 0.f;
    for (int b = 0; b < B; ++b)
        acc += att[r * B + b] * basis[(size_t)b * FE + fe];
    W[idx] = acc;
}

// ---------------------------------------------------------------------------
// Zero fill (float4 grid-stride)
// ---------------------------------------------------------------------------
__global__ void zero_kernel(float4* __restrict__ p, long long n4) {
    long long i = (long long)blockIdx.x * blockDim.x + threadIdx.x;
    long long stride = (long long)gridDim.x * blockDim.x;
    float4 z = make_float4(0.f, 0.f, 0.f, 0.f);
    for (; i < n4; i += stride) p[i] = z;
}

// ---------------------------------------------------------------------------
// h[r, node, :] = (feat[node, :] @ W[r]) * cj[node]
// One workgroup: 128-node x 128-col tile for rating r = blockIdx.y.
// 8 waves; wave w owns a 32-row x 64-col sub-tile (acc[2][4] of v8f).
// WMMA f32 16x16x4; per k-step: 2 A-frag + 4 B-frag ds_load_b64, 8 WMMAs.
//   sA : async GLOBAL_LOAD_ASYNC_TO_LDS_B128 fill (ASYNCcnt), stride 36
//   sBp: K packed in pairs, sBp[k/2][2*col + (k&1)], row stride 288
//        (288 mod 64 == 32 -> lane halves hit disjoint bank sets)
// ---------------------------------------------------------------------------
__global__ __launch_bounds__(256)
void transform_kernel(const float* __restrict__ feat,
                      const float* __restrict__ cj,
                      const float* __restrict__ W,
                      float* __restrict__ h,
                      int N, int R) {
    __shared__ float sA[128][SA_STRIDE];                 // 128 x 36
    __shared__ float sBp[KT / 2][2 * EFF_DIM + 32];      // 16 x 288

    const int r = blockIdx.y;
    const int nodeBase = blockIdx.x * 128;
    const int tid  = threadIdx.x;
    const int wave = tid >> 5;
    const int lane = tid & 31;
    const int lh = lane & 15;            // low-half lane index
    const int hi = lane >> 4;            // 0 or 1
    const int kk = hi * 2;               // K sub-offset per ISA 16x4 layout

    const int rw = (wave & 3) * 32;      // wave row strip
    const int cw = (wave >> 2) * 64;     // wave col strip

    v8f acc[2][4] = {};

    const float* Wr = W + (size_t)r * F_DIM * EFF_DIM;
    const bool tail = (nodeBase + 128 > N);

    // zero OOB rows once (they are never overwritten by the tile loads)
    if (tail) {
        for (int i = tid; i < 128 * KT; i += 256) {
            int row = i / KT, col = i - row * KT;
            if (nodeBase + row >= N) sA[row][col] = 0.f;
        }
        __syncthreads();
    }

    for (int kb = 0; kb < F_DIM; kb += KT) {
        // stage feat tile 128 x KT
#if HAS_ASYNC_LDS
        // 1024 b128 segments (8 per row), 4 per thread, register-file bypass
        for (int i = tid; i < 128 * (KT / 4); i += 256) {
            int row = i >> 3, seg = i & 7;
            int node = nodeBase + row;
            if (node < N) {
                __builtin_amdgcn_global_load_async_to_lds_b128(
                    (as1_i4p)(const_cast<float*>(feat) +
                              (size_t)node * F_DIM + kb + seg * 4),
                    (as3_i4p)&sA[row][seg * 4], 0, 0);
            }
        }
#else
        for (int i = tid; i < 128 * KT; i += 256) {
            int row = i / KT, col = i - row * KT;
            int node = nodeBase + row;
            if (node < N) sA[row][col] = feat[(size_t)node * F_DIM + kb + col];
        }
#endif
        // cooperative load: W chunk KT x 128, K-pair packed
        for (int i = tid; i < KT * EFF_DIM; i += 256) {
            int row = i / EFF_DIM, col = i - row * EFF_DIM;
            sBp[row >> 1][2 * col + (row & 1)] = Wr[(size_t)(kb + row) * EFF_DIM + col];
        }
#if HAS_ASYNC_LDS
        async_wait0();
#endif
        __syncthreads();

        #pragma unroll
        for (int k = 0; k < KT; k += 4) {
            const int kh = (k >> 1) + hi;            // packed B row = (k+kk)/2
            v2f a0 = *(const v2f*)&sA[rw + lh][k + kk];
            v2f a1 = *(const v2f*)&sA[rw + 16 + lh][k + kk];
            #pragma unroll
            for (int ct = 0; ct < 4; ++ct) {
                v2f b = *(const v2f*)&sBp[kh][2 * (cw + ct * 16 + lh)];
                acc[0][ct] = __builtin_amdgcn_wmma_f32_16x16x4_f32(
                    false, a0, false, b, (short)0, acc[0][ct], false, false);
                acc[1][ct] = __builtin_amdgcn_wmma_f32_16x16x4_f32(
                    false, a1, false, b, (short)0, acc[1][ct], false, false);
            }
        }
        __syncthreads();
    }

    // store with cj scaling; D layout: VGPR j -> row j + 8*hi, col = lane&15
    #pragma unroll
    for (int ri = 0; ri < 2; ++ri) {
        #pragma unroll
        for (int ct = 0; ct < 4; ++ct) {
            int col = cw + ct * 16 + lh;
            #pragma unroll
            for (int j = 0; j < 8; ++j) {
                int m = rw + ri * 16 + j + 8 * hi;
                int node = nodeBase + m;
                if (node < N)
                    h[((size_t)r * N + node) * EFF_DIM + col] = acc[ri][ct][j] * cj[node];
            }
        }
    }
}

// ---------------------------------------------------------------------------
// Edge scatter: one wave32 per edge; lane handles 4 consecutive floats.
// z[r, dst, :] += h[r, src, :]
// ---------------------------------------------------------------------------
__device__ __forceinline__ void atomic_add_f32(float* p, float v) {
    __hip_atomic_fetch_add(p, v, __ATOMIC_RELAXED, __HIP_MEMORY_SCOPE_AGENT);
}

__global__ __launch_bounds__(256)
void scatter_kernel(const float* __restrict__ h,
                    float* __restrict__ z,
                    const int* __restrict__ esrc,
                    const int* __restrict__ edst,
                    long long E, int R, int Nsrc, int Ndst) {
    long long warp = (long long)blockIdx.x * 8 + (threadIdx.x >> 5);
    long long total = (long long)R * E;
    if (warp >= total) return;
    const int lane = threadIdx.x & 31;
    int r = (int)(warp / E);
    long long e = warp - (long long)r * E;
    int src = esrc[(size_t)r * E + e];
    int dst = edst[(size_t)r * E + e];
    const float4 v = *(const float4*)(h + ((size_t)r * Nsrc + src) * EFF_DIM + lane * 4);
    float* zp = z + ((size_t)r * Ndst + dst) * EFF_DIM + lane * 4;
    atomic_add_f32(zp + 0, v.x);
    atomic_add_f32(zp + 1, v.y);
    atomic_add_f32(zp + 2, v.z);
    atomic_add_f32(zp + 3, v.w);
}

// ---------------------------------------------------------------------------
// out[n, r, :] = (z[r, n, :] * ci[n]) @ fc_w + fc_b
// Row index m = n*R + r maps 1:1 to the output layout.
// 8 waves/block, wave handles 32 rows x 64 cols (acc[2][4] of v8f).
// fc_w lives in LDS in K-pair-packed form (single b64 per fragment).
// ---------------------------------------------------------------------------
__global__ __launch_bounds__(256)
void project_kernel(const float* __restrict__ z,
                    const float* __restrict__ ci,
                    const float* __restrict__ fc_w,
                    const float* __restrict__ fc_b,
                    float* __restrict__ out,
                    int N, int R) {
    __shared__ float sWp[EFF_DIM / 2][2 * OUT_DIM + 32];   // 64 x 160 floats

    const int tid  = threadIdx.x;
    const int wave = tid >> 5;
    const int lane = tid & 31;
    const int lh = lane & 15;
    const int hi = lane >> 4;
    const int kk = hi * 2;

    for (int i = tid; i < EFF_DIM * OUT_DIM; i += 256) {
        int row = i / OUT_DIM, col = i - row * OUT_DIM;
        sWp[row >> 1][2 * col + (row & 1)] = fc_w[(size_t)row * OUT_DIM + col];
    }
    __syncthreads();

    const long long totalRows = (long long)N * R;
    const long long rowBase = ((long long)blockIdx.x * 8 + wave) * 32;
    if (rowBase >= totalRows) return;

    // per-lane A-row pointers for the two 16-row tiles (clamped for tail)
    long long rA0 = rowBase + lh;        if (rA0 >= totalRows) rA0 = totalRows - 1;
    long long rA1 = rowBase + 16 + lh;   if (rA1 >= totalRows) rA1 = totalRows - 1;
    int n0 = (int)(rA0 / R), r0 = (int)(rA0 - (long long)n0 * R);
    int n1 = (int)(rA1 / R), r1 = (int)(rA1 - (long long)n1 * R);
    const float* arow0 = z + ((size_t)r0 * N + n0) * EFF_DIM;
    const float* arow1 = z + ((size_t)r1 * N + n1) * EFF_DIM;
    const float scale0 = ci[n0];
    const float scale1 = ci[n1];

    v8f acc[2][4] = {};
    #pragma unroll
    for (int k = 0; k < EFF_DIM; k += 4) {
        const int kh = (k >> 1) + hi;
        v2f ga0 = *(const v2f*)&arow0[k + kk];
        v2f ga1 = *(const v2f*)&arow1[k + kk];
        v2f a0 = { ga0.x * scale0, ga0.y * scale0 };
        v2f a1 = { ga1.x * scale1, ga1.y * scale1 };
        #pragma unroll
        for (int ct = 0; ct < 4; ++ct) {
            v2f b = *(const v2f*)&sWp[kh][2 * (ct * 16 + lh)];
            acc[0][ct] = __builtin_amdgcn_wmma_f32_16x16x4_f32(
                false, a0, false, b, (short)0, acc[0][ct], false, false);
            acc[1][ct] = __builtin_amdgcn_wmma_f32_16x16x4_f32(
                false, a1, false, b, (short)0, acc[1][ct], false, false);
        }
    }

    #pragma unroll
    for (int ri = 0; ri < 2; ++ri) {
        #pragma unroll
        for (int ct = 0; ct < 4; ++ct) {
            int col = ct * 16 + lh;
            float bias = fc_b[col];
            #pragma unroll
            for (int j = 0; j < 8; ++j) {
                long long m = rowBase + ri * 16 + j + 8 * hi;
                if (m < totalRows)
                    out[(size_t)m * OUT_DIM + col] = acc[ri][ct][j] + bias;
            }
        }
    }
}

// ---------------------------------------------------------------------------
extern "C" void kernel_launch(void* const* d_in, const int* in_sizes, int n_in,
                              void* d_out, int out_size, void* d_ws, size_t ws_size,
                              hipStream_t stream) {
    const float* drug_feat = (const float*)d_in[0];
    const float* dis_feat  = (const float*)d_in[1];
    const float* cj_drug   = (const float*)d_in[2];
    const float* ci_drug   = (const float*)d_in[3];
    const float* cj_dis    = (const float*)d_in[4];
    const float* ci_dis    = (const float*)d_in[5];
    const float* att       = (const float*)d_in[6];
    const float* basis     = (const float*)d_in[7];
    const float* fc_w      = (const float*)d_in[8];
    const float* fc_b      = (const float*)d_in[9];
    const int*   edge_drug = (const int*)d_in[10];
    const int*   edge_dis  = (const int*)d_in[11];

    const int Ndrug = in_sizes[2];
    const int Ndis  = in_sizes[4];
    const int FE    = F_DIM * EFF_DIM;
    const int B     = in_sizes[7] / FE;          // basis [B, F, EFF]
    const int R     = in_sizes[6] / B;           // att   [R, B]
    const long long E = (long long)in_sizes[10] / R;

    // workspace layout (floats)
    float* W      = (float*)d_ws;
    float* h_drug = W      + (size_t)R * FE;
    float* h_dis  = h_drug + (size_t)R * Ndrug * EFF_DIM;
    float* z_dis  = h_dis  + (size_t)R * Ndis  * EFF_DIM;
    float* z_drug = z_dis  + (size_t)R * Ndis  * EFF_DIM;

    float* out_drug = (float*)d_out;
    float* out_dis  = out_drug + (size_t)Ndrug * R * OUT_DIM;

    // 1. build W
    {
        int total = R * FE;
        build_w_kernel<<<(total + 255) / 256, 256, 0, stream>>>(att, basis, W, R, B, FE);
    }
    // 2. zero z accumulators (z_dis and z_drug are contiguous)
    {
        long long zfloats = (long long)R * (Ndis + Ndrug) * EFF_DIM;
        long long n4 = zfloats / 4;
        int blocks = (int)((n4 + 255) / 256);
        if (blocks > 65535) blocks = 65535;
        zero_kernel<<<blocks, 256, 0, stream>>>((float4*)z_dis, n4);
    }
    // 3. per-rating transforms (WMMA f32, async LDS staging)
    {
        dim3 gridD((Ndrug + 127) / 128, R);
        transform_kernel<<<gridD, 256, 0, stream>>>(drug_feat, cj_drug, W, h_drug, Ndrug, R);
        dim3 gridI((Ndis + 127) / 128, R);
        transform_kernel<<<gridI, 256, 0, stream>>>(dis_feat, cj_dis, W, h_dis, Ndis, R);
    }
    // 4. edge scatter (fp32 atomics)
    {
        long long warps = (long long)R * E;
        int blocks = (int)((warps + 7) / 8);
        scatter_kernel<<<blocks, 256, 0, stream>>>(h_drug, z_dis, edge_drug, edge_dis,
                                                   E, R, Ndrug, Ndis);
        scatter_kernel<<<blocks, 256, 0, stream>>>(h_dis, z_drug, edge_dis, edge_drug,
                                                   E, R, Ndis, Ndrug);
    }
    // 5. output projection (WMMA f32, fc_w in LDS)
    {
        long long rowsD = (long long)Ndrug * R;
        int blocksD = (int)((rowsD + 255) / 256);
        project_kernel<<<blocksD, 256, 0, stream>>>(z_drug, ci_drug, fc_w, fc_b,
                                                    out_drug, Ndrug, R);
        long long rowsI = (long long)Ndis * R;
        int blocksI = (int)((rowsI + 255) / 256);
        project_kernel<<<blocksI, 256, 0, stream>>>(z_dis, ci_dis, fc_w, fc_b,
                                                    out_dis, Ndis, R);
    }
}